// EfficientGlobalPointer_25494925869454
// MI455X (gfx1250) — compile-verified
//
#include <hip/hip_runtime.h>
#include <hip/hip_bf16.h>

typedef __attribute__((ext_vector_type(2))) float v2f;
typedef __attribute__((ext_vector_type(8))) float v8f;

#define BB 16
#define LL 512
#define HH 1024
#define DD 64      // HEAD_SIZE
#define NH 12      // HEADS
#define INFV 1000000000000.0f

// ---------------------------------------------------------------------------
// Kernel 1: x = inputs @ W1 + b1   (M=8192, N=128, K=1024), f32 WMMA 16x16x4.
// One wave computes one 16x16 tile. A (inputs rows): lane reads contiguous
// b64 at K = k0 + 2*(lane>=16). B (W1): lane ln holds column n0+ln, VGPR v
// holds K = k0 + v + 2*(lane>=16).
// ---------------------------------------------------------------------------
__global__ void egp_proj_kernel(const float* __restrict__ inputs,
                                const float* __restrict__ W1,
                                const float* __restrict__ b1,
                                float* __restrict__ x) {
    const int wave = threadIdx.x >> 5;
    const int lane = threadIdx.x & 31;
    const int tile = blockIdx.x * 8 + wave;      // 4096 tiles = 512 mt x 8 nt
    const int mt = tile >> 3;
    const int nt = tile & 7;
    const int row0 = mt * 16;
    const int n0 = nt * 16;
    const int ln = lane & 15;
    const int hi = lane >> 4;

    const float* arow = inputs + (size_t)(row0 + ln) * HH + 2 * hi;
    const float* bcol = W1 + n0 + ln;

    v8f acc = {};
    #pragma unroll 4
    for (int k0 = 0; k0 < HH; k0 += 4) {
        v2f a = *(const v2f*)(arow + k0);
        v2f b;
        b.x = bcol[(size_t)(k0 + 2 * hi)     * 128];
        b.y = bcol[(size_t)(k0 + 2 * hi + 1) * 128];
        acc = __builtin_amdgcn_wmma_f32_16x16x4_f32(
            false, a, false, b, (short)0, acc, false, false);
    }

    const float badd = b1[n0 + ln];
    float* o = x + (size_t)row0 * 128 + n0 + ln;
    #pragma unroll
    for (int v = 0; v < 8; ++v)
        o[(size_t)(v + 8 * hi) * 128] = acc[v] + badd;
}

// ---------------------------------------------------------------------------
// Kernel 2: RoPE. q = x[...,0::2], k = x[...,1::2]; rotary pair i mixes
// x cols {4i, 4i+2} (q) and {4i+1, 4i+3} (k). Outputs stored TRANSPOSED as
// qT/kT [b, d, l] so the qk^T WMMA loads are coalesced per half-wave.
// ---------------------------------------------------------------------------
__global__ void egp_rope_kernel(const float* __restrict__ x,
                                float* __restrict__ qT,
                                float* __restrict__ kT) {
    const int idx   = blockIdx.x;          // B * 32 * 2 = 1024 blocks
    const int lhalf = idx & 1;
    const int i     = (idx >> 1) & 31;     // rotary pair index
    const int b     = idx >> 6;
    const int l     = lhalf * 256 + threadIdx.x;

    const float* xr = x + (size_t)(b * LL + l) * 128 + 4 * i;
    const float q1 = xr[0], k1 = xr[1], q2 = xr[2], k2 = xr[3];

    const float inv = __powf(10000.0f, -(float)(2 * i) / 64.0f);
    float s, c;
    __sincosf((float)l * inv, &s, &c);

    const size_t base = ((size_t)b * DD + 2 * i) * LL + l;
    qT[base]      = q1 * c - q2 * s;
    qT[base + LL] = q1 * s + q2 * c;
    kT[base]      = k1 * c - k2 * s;
    kT[base + LL] = k1 * s + k2 * c;
}

// ---------------------------------------------------------------------------
// Kernel 3: biasT[b, j, l] = (x[b,l,:] @ W2[:,j] + b2[j]) * 0.5  (K=128, N=24)
// ---------------------------------------------------------------------------
__global__ void egp_bias_kernel(const float* __restrict__ x,
                                const float* __restrict__ W2,
                                const float* __restrict__ b2,
                                float* __restrict__ biasT) {
    const int idx   = blockIdx.x;          // B * 24 * 2 = 768 blocks
    const int lhalf = idx & 1;
    const int j     = (idx >> 1) % 24;
    const int b     = idx / 48;
    const int l     = lhalf * 256 + threadIdx.x;

    const float* xr = x + (size_t)(b * LL + l) * 128;
    float acc = b2[j];
    #pragma unroll 8
    for (int k = 0; k < 128; ++k)
        acc += xr[k] * W2[k * 24 + j];
    biasT[((size_t)b * 24 + j) * LL + l] = acc * 0.5f;
}

// ---------------------------------------------------------------------------
// Kernel 4: main output. One wave per 16x16 qk tile (K=64 via 16 WMMA steps),
// then fan out to 12 heads with rank-1 bias adds + masks, writing 201 MB.
// C/D layout: lane -> N = n0 + (lane&15); VGPR v -> M = m0 + v + 8*(lane>=16).
// ---------------------------------------------------------------------------
__global__ void egp_logits_kernel(const float* __restrict__ qT,
                                  const float* __restrict__ kT,
                                  const float* __restrict__ biasT,
                                  const float* __restrict__ am,
                                  float* __restrict__ out) {
    const int wave = threadIdx.x >> 5;
    const int lane = threadIdx.x & 31;
    const int tile = blockIdx.x * 8 + wave;   // 16384 tiles: b x mt x nt
    const int nt = tile & 31;
    const int mt = (tile >> 5) & 31;
    const int b  = tile >> 10;
    const int m0 = mt * 16;
    const int n0 = nt * 16;
    const int ln = lane & 15;
    const int hi = lane >> 4;

    const float* qb = qT + (size_t)b * DD * LL + m0 + ln;
    const float* kb = kT + (size_t)b * DD * LL + n0 + ln;

    v8f acc = {};
    #pragma unroll 4
    for (int k0 = 0; k0 < DD; k0 += 4) {
        const int kk = k0 + 2 * hi;
        v2f a, bm;
        a.x  = qb[(size_t)kk * LL];
        a.y  = qb[(size_t)(kk + 1) * LL];
        bm.x = kb[(size_t)kk * LL];
        bm.y = kb[(size_t)(kk + 1) * LL];
        acc = __builtin_amdgcn_wmma_f32_16x16x4_f32(
            false, a, false, bm, (short)0, acc, false, false);
    }

    const int n = n0 + ln;
    const float amn = am[b * LL + n];
    float pre[8];
    #pragma unroll
    for (int v = 0; v < 8; ++v) {
        const int m = m0 + v + 8 * hi;
        const float amm = am[b * LL + m];
        const float maskv = (1.0f - amn * amm) * INFV + ((m > n) ? INFV : 0.0f);
        pre[v] = acc[v] * 0.125f - maskv;   // 1/sqrt(64) = 0.125
    }

    for (int h = 0; h < NH; ++h) {
        const float b0 = biasT[((size_t)b * 24 + 2 * h) * LL + n];       // per-col
        const float* b1row = biasT + ((size_t)b * 24 + 2 * h + 1) * LL;  // per-row
        float* obase = out + ((size_t)(b * NH + h) * LL) * LL + n;
        #pragma unroll
        for (int v = 0; v < 8; ++v) {
            const int m = m0 + v + 8 * hi;
            obase[(size_t)m * LL] = pre[v] + b0 + b1row[m];
        }
    }
}

extern "C" void kernel_launch(void* const* d_in, const int* in_sizes, int n_in,
                              void* d_out, int out_size, void* d_ws, size_t ws_size,
                              hipStream_t stream) {
    const float* inputs = (const float*)d_in[0];   // (16, 512, 1024)
    const float* amask  = (const float*)d_in[1];   // (16, 512)
    const float* W1     = (const float*)d_in[2];   // (1024, 128)
    const float* b1     = (const float*)d_in[3];   // (128,)
    const float* W2     = (const float*)d_in[4];   // (128, 24)
    const float* b2     = (const float*)d_in[5];   // (24,)
    float* out = (float*)d_out;                    // (16, 12, 512, 512)

    float* x     = (float*)d_ws;                   // 8192 x 128   (4 MB)
    float* qT    = x  + (size_t)BB * LL * 128;     // 16 x 64 x 512 (2 MB)
    float* kT    = qT + (size_t)BB * DD * LL;      // 16 x 64 x 512 (2 MB)
    float* biasT = kT + (size_t)BB * DD * LL;      // 16 x 24 x 512 (0.79 MB)

    egp_proj_kernel  <<<512,  256, 0, stream>>>(inputs, W1, b1, x);
    egp_rope_kernel  <<<1024, 256, 0, stream>>>(x, qT, kT);
    egp_bias_kernel  <<<768,  256, 0, stream>>>(x, W2, b2, biasT);
    egp_logits_kernel<<<2048, 256, 0, stream>>>(qT, kT, biasT, amask, out);
}